// SSMCausalConvBlock_49624052138250
// MI455X (gfx1250) — compile-verified
//
#include <hip/hip_runtime.h>
#include <cmath>

// Problem constants (from reference)
#define B_  4
#define L_  4096
#define D_  128
#define N_  32
#define BL  (B_ * L_)        // 16384 rows
#define XD  (2 * N_ + D_)    // 192
#define CHUNKS 32
#define CHLEN  (L_ / CHUNKS) // 128

typedef __attribute__((ext_vector_type(2))) float v2f;
typedef __attribute__((ext_vector_type(8))) float v8f;

// Native fp32 WMMA: D(16x16,f32) = A(16x4,f32) x B(4x16,f32) + C
__device__ __forceinline__ v8f wmma4(v2f a, v2f b, v8f c) {
  return __builtin_amdgcn_wmma_f32_16x16x4_f32(false, a, false, b, (short)0, c,
                                               false, false);
}

__device__ __forceinline__ float clampf(float x, float lo, float hi) {
  return fminf(fmaxf(x, lo), hi);
}
__device__ __forceinline__ float softplusf(float x) {
  // numerically-stable softplus: max(x,0) + log1p(exp(-|x|))
  return fmaxf(x, 0.f) + log1pf(expf(-fabsf(x)));
}

// 16x16 f32 tile accumulate over K in steps of 4 (A rows from `arow` with
// row stride `lda` floats; B from `w` with row stride `ldb` floats).
__device__ __forceinline__ v8f gemm_tile_k128(const float* __restrict__ arow,
                                              int lda, const float* __restrict__ w,
                                              int ldb, int n0, int m, int kh,
                                              v8f acc) {
#pragma unroll 4
  for (int kt = 0; kt < 32; ++kt) {
    const int k = kt * 4 + 2 * kh;
    v2f a = *(const v2f*)(arow + (size_t)m * lda + k);
    v2f b;
    b.x = w[(size_t)k * ldb + n0 + m];
    b.y = w[(size_t)(k + 1) * ldb + n0 + m];
    acc = wmma4(a, b, acc);
  }
  return acc;
}

// ---------------------------------------------------------------------------
// Kernel 1: fused x@W_xp (192 cols) -> delta_in (LDS) / Bm / Cm; then
// delta = softplus(delta_in @ W_dt + b_dt).  One wave per 16-row tile.
// nt ranges split statically so every store path is branch-free.
// ---------------------------------------------------------------------------
__global__ __launch_bounds__(32)
void k_proj(const float* __restrict__ x, const float* __restrict__ W_xp,
            const float* __restrict__ W_dt, const float* __restrict__ b_dt,
            float* __restrict__ delta, float* __restrict__ Bm,
            float* __restrict__ Cm) {
  __shared__ float lds[16 * D_];          // delta_in tile (16 x 128)
  const int lane = threadIdx.x;
  const int m = lane & 15;                // row-in-tile (A) / col-in-tile (B,D)
  const int kh = lane >> 4;               // K half-select
  const int r0 = blockIdx.x * 16;
  const float* xrow = x + (size_t)r0 * D_;

  // GEMM1a: delta_in columns (0..127) -> LDS
  for (int nt = 0; nt < 8; ++nt) {
    v8f acc = {};
    acc = gemm_tile_k128(xrow, D_, W_xp, XD, nt * 16, m, kh, acc);
#pragma unroll
    for (int j = 0; j < 8; ++j)
      lds[(j + 8 * kh) * D_ + nt * 16 + m] = acc[j];
  }
  // GEMM1b: Bm columns (128..159)
  for (int nt = 0; nt < 2; ++nt) {
    v8f acc = {};
    acc = gemm_tile_k128(xrow, D_, W_xp, XD, 128 + nt * 16, m, kh, acc);
#pragma unroll
    for (int j = 0; j < 8; ++j)
      Bm[(size_t)(r0 + j + 8 * kh) * N_ + nt * 16 + m] = acc[j];
  }
  // GEMM1c: Cm columns (160..191)
  for (int nt = 0; nt < 2; ++nt) {
    v8f acc = {};
    acc = gemm_tile_k128(xrow, D_, W_xp, XD, 160 + nt * 16, m, kh, acc);
#pragma unroll
    for (int j = 0; j < 8; ++j)
      Cm[(size_t)(r0 + j + 8 * kh) * N_ + nt * 16 + m] = acc[j];
  }
  __syncthreads();

  // GEMM2: delta = softplus(delta_in @ W_dt + b_dt)
  for (int nt = 0; nt < 8; ++nt) {
    v8f acc = {};
    acc = gemm_tile_k128(lds, D_, W_dt, D_, nt * 16, m, kh, acc);
#pragma unroll
    for (int j = 0; j < 8; ++j) {
      const int row = j + 8 * kh;
      const int col = nt * 16 + m;
      delta[(size_t)(r0 + row) * D_ + col] = softplusf(acc[j] + b_dt[col]);
    }
  }
}

// ---------------------------------------------------------------------------
// Kernel 2a: per-(chain,chunk) partial sums of dA (cumsum is associative; the
// clip applies only when *reading* the prefix).  Wave per (b,d,chunk), lanes=n.
// ---------------------------------------------------------------------------
__global__ __launch_bounds__(32)
void k_scan_partial(const float* __restrict__ delta,
                    const float* __restrict__ A_log,
                    float* __restrict__ part) {
  const int blk = blockIdx.x;
  const int bd = blk >> 5, chunk = blk & 31;
  const int b = bd >> 7, d = bd & 127;
  const int n = threadIdx.x;
  const float A_dn = -expf(A_log[d * N_ + n]);
  const float* dp = delta + ((size_t)b * L_ + chunk * CHLEN) * D_ + d;
  float acc = 0.f;
  for (int i = 0; i < CHLEN; ++i)
    acc += clampf(dp[(size_t)i * D_] * A_dn, -5.f, 5.f);
  part[((size_t)bd * N_ + n) * CHUNKS + chunk] = acc;   // [bd][n][chunk]
}

// ---------------------------------------------------------------------------
// Kernel 2b: exclusive scan of the 32 chunk totals per chain (wave32 shfl).
// ---------------------------------------------------------------------------
__global__ __launch_bounds__(32)
void k_scan_offsets(float* __restrict__ part) {
  const int c = blockIdx.x;          // chain id = bd*32 + n
  const int lane = threadIdx.x;      // chunk
  const float v = part[(size_t)c * CHUNKS + lane];
  float s = v;
#pragma unroll
  for (int off = 1; off < 32; off <<= 1) {
    float t = __shfl_up(s, off, 32);
    if (lane >= off) s += t;
  }
  part[(size_t)c * CHUNKS + lane] = s - v;   // exclusive prefix
}

// ---------------------------------------------------------------------------
// Kernel 2c: replay chunk seeded with offset, gate with exp(clip(prefix)),
// contract over n via 32-lane shuffle reduce, emit h_t.
// ---------------------------------------------------------------------------
__global__ __launch_bounds__(32)
void k_scan_apply(const float* __restrict__ delta,
                  const float* __restrict__ h_prev,
                  const float* __restrict__ Bm, const float* __restrict__ Cm,
                  const float* __restrict__ A_log,
                  const float* __restrict__ D_param,
                  const float* __restrict__ part, float* __restrict__ h_t) {
  const int blk = blockIdx.x;
  const int bd = blk >> 5, chunk = blk & 31;
  const int b = bd >> 7, d = bd & 127;
  const int n = threadIdx.x;
  const float A_dn = -expf(A_log[d * N_ + n]);
  const float Dp = D_param[d];
  float acc = part[((size_t)bd * N_ + n) * CHUNKS + chunk];
  const int l0 = chunk * CHLEN;
  for (int i = 0; i < CHLEN; ++i) {
    const size_t r = (size_t)b * L_ + l0 + i;
    const float dv = delta[r * D_ + d];
    const float hp = h_prev[r * D_ + d];
    const float Bv = Bm[r * N_ + n];
    const float Cv = Cm[r * N_ + n];
    __builtin_prefetch(&Bm[(r + 4) * N_ + n], 0, 0);   // L2-resident streams
    __builtin_prefetch(&Cm[(r + 4) * N_ + n], 0, 0);
    const float dA = clampf(dv * A_dn, -5.f, 5.f);
    const float g = expf(clampf(acc, -10.f, 10.f));    // exclusive prefix gate
    float contrib = dv * hp * Bv * g * Cv;
#pragma unroll
    for (int off = 16; off > 0; off >>= 1)
      contrib += __shfl_xor(contrib, off, 32);
    if (n == 0)
      h_t[r * D_ + d] = clampf(contrib * (1.f + Dp), -100.f, 100.f);
    acc += dA;
  }
}

// ---------------------------------------------------------------------------
// Kernel 3: ssm_out = relu(h_t @ W_lf + b_lf), WMMA, one wave per 16 rows.
// ---------------------------------------------------------------------------
__global__ __launch_bounds__(32)
void k_outproj(const float* __restrict__ h_t, const float* __restrict__ W_lf,
               const float* __restrict__ b_lf, float* __restrict__ ssm) {
  const int lane = threadIdx.x;
  const int m = lane & 15, kh = lane >> 4;
  const int r0 = blockIdx.x * 16;
  const float* arow = h_t + (size_t)r0 * D_;
  for (int nt = 0; nt < 8; ++nt) {
    v8f acc = {};
    acc = gemm_tile_k128(arow, D_, W_lf, D_, nt * 16, m, kh, acc);
#pragma unroll
    for (int j = 0; j < 8; ++j) {
      const int row = j + 8 * kh, col = nt * 16 + m;
      ssm[(size_t)(r0 + row) * D_ + col] = fmaxf(acc[j] + b_lf[col], 0.f);
    }
  }
}

// ---------------------------------------------------------------------------
// Kernel 4: causal conv (k=3, left pad 2) as a K=384 WMMA GEMM over shifted A
// rows, then relu + gate.  Boundary guard hoisted to a uniform per-block
// branch: only the first 16-row tile of each batch can under-run (l0 == 0),
// so 1020/1024 blocks run the guard-free inner loop.
// ---------------------------------------------------------------------------
__global__ __launch_bounds__(32)
void k_conv_fuse(const float* __restrict__ ssm, const float* __restrict__ ck,
                 const float* __restrict__ conv_b, float* __restrict__ fused) {
  const int lane = threadIdx.x;
  const int m = lane & 15, kh = lane >> 4;
  const int r0 = blockIdx.x * 16;
  const int b = r0 >> 12;         // / L_
  const int l0 = r0 & (L_ - 1);
  const float* base = ssm + (size_t)b * L_ * D_;

  for (int nt = 0; nt < 8; ++nt) {
    v8f acc = {};
    const int n0 = nt * 16;
    if (l0 != 0) {
      // Interior tile: lr = l0+m+w-2 >= 14, no guards needed.
#pragma unroll
      for (int w = 0; w < 3; ++w)
        acc = gemm_tile_k128(base + (size_t)(l0 + w - 2) * D_, D_,
                             ck + (size_t)w * D_ * D_, D_, n0, m, kh, acc);
    } else {
      // First tile of the batch: rows m+w-2 < 0 read the causal zero-pad.
#pragma unroll
      for (int w = 0; w < 3; ++w) {
        const int lr = m + w - 2;
        const bool ok = (lr >= 0);
        const float* arow = base + (size_t)(ok ? lr : 0) * D_;
        const float* kw = ck + (size_t)w * D_ * D_;
        for (int kt = 0; kt < 32; ++kt) {
          const int k = kt * 4 + 2 * kh;
          v2f a = {};
          if (ok) a = *(const v2f*)(arow + k);
          v2f bb;
          bb.x = kw[(size_t)k * D_ + n0 + m];
          bb.y = kw[(size_t)(k + 1) * D_ + n0 + m];
          acc = wmma4(a, bb, acc);
        }
      }
    }
#pragma unroll
    for (int j = 0; j < 8; ++j) {
      const int row = j + 8 * kh, col = n0 + m;
      const size_t r = (size_t)r0 + row;
      const float s = ssm[r * D_ + col];
      const float c = fmaxf(acc[j] + conv_b[col], 0.f);
      fused[r * D_ + col] = s * c;
    }
  }
}

// ---------------------------------------------------------------------------
extern "C" void kernel_launch(void* const* d_in, const int* in_sizes, int n_in,
                              void* d_out, int out_size, void* d_ws,
                              size_t ws_size, hipStream_t stream) {
  const float* x       = (const float*)d_in[0];
  const float* h_prev  = (const float*)d_in[1];
  const float* A_log   = (const float*)d_in[2];
  const float* D_param = (const float*)d_in[3];
  const float* W_xp    = (const float*)d_in[4];
  const float* W_dt    = (const float*)d_in[5];
  const float* b_dt    = (const float*)d_in[6];
  const float* W_lf    = (const float*)d_in[7];
  const float* b_lf    = (const float*)d_in[8];
  const float* conv_k  = (const float*)d_in[9];
  const float* conv_b  = (const float*)d_in[10];

  float* fused = (float*)d_out;                 // [BL, D]
  float* h_t   = fused + (size_t)BL * D_;       // [BL, D] (second output)

  float* ws    = (float*)d_ws;
  float* delta = ws;                            // BL*D
  float* Bm    = delta + (size_t)BL * D_;       // BL*N
  float* Cm    = Bm + (size_t)BL * N_;          // BL*N
  float* ssm   = Cm + (size_t)BL * N_;          // BL*D
  float* part  = ssm + (size_t)BL * D_;         // B*D*N*CHUNKS

  k_proj<<<BL / 16, 32, 0, stream>>>(x, W_xp, W_dt, b_dt, delta, Bm, Cm);
  k_scan_partial<<<B_ * D_ * CHUNKS, 32, 0, stream>>>(delta, A_log, part);
  k_scan_offsets<<<B_ * D_ * N_, 32, 0, stream>>>(part);
  k_scan_apply<<<B_ * D_ * CHUNKS, 32, 0, stream>>>(delta, h_prev, Bm, Cm,
                                                    A_log, D_param, part, h_t);
  k_outproj<<<BL / 16, 32, 0, stream>>>(h_t, W_lf, b_lf, ssm);
  k_conv_fuse<<<BL / 16, 32, 0, stream>>>(ssm, conv_k, conv_b, fused);
}